// SEBlock_15771119911511
// MI455X (gfx1250) — compile-verified
//
#include <hip/hip_runtime.h>
#include <math.h>

// ---------------------------------------------------------------------------
// SE block for MI455X (gfx1250, wave32).
//   x:[64,512,4096] f32 -> mean over T -> 2 tiny GEMMs (WMMA f32 16x16x4)
//   -> sigmoid channel weights -> rescale x.
// Bandwidth-bound: ~1.5 GB HBM traffic @ 23.3 TB/s => ~65us floor.
// ---------------------------------------------------------------------------

typedef float f4  __attribute__((ext_vector_type(4)));
typedef float v2f __attribute__((ext_vector_type(2)));
typedef float v8f __attribute__((ext_vector_type(8)));

#define BB 64
#define CC 512
#define TT 4096
#define HH 64

// ---------------------------------------------------------------------------
// Pass 1: per-(b,c) mean over T.  One block per row, 256 threads,
// 4x float4 (global_load_b128, non-temporal) per thread, wave32 shuffle tree.
// ---------------------------------------------------------------------------
__global__ __launch_bounds__(256) void se_mean(const float* __restrict__ x,
                                               float* __restrict__ sq) {
  const int row = blockIdx.x;                       // b*C + c
  const f4* xr = (const f4*)(x + (size_t)row * TT); // 1024 float4 per row
  const int tid = threadIdx.x;

  float s = 0.f;
#pragma unroll
  for (int i = 0; i < 4; ++i) {
    f4 v = __builtin_nontemporal_load(&xr[tid + i * 256]);
    s += v.x + v.y + v.z + v.w;
  }

  // wave32 reduction
#pragma unroll
  for (int off = 16; off > 0; off >>= 1) s += __shfl_down(s, off, 32);

  __shared__ float red[8];
  const int lane = tid & 31, wid = tid >> 5;
  if (lane == 0) red[wid] = s;
  __syncthreads();
  if (tid < 8) {
    float t = red[tid];
#pragma unroll
    for (int off = 4; off > 0; off >>= 1) t += __shfl_down(t, off, 8);
    if (tid == 0) sq[row] = t * (1.0f / (float)TT);
  }
}

// ---------------------------------------------------------------------------
// Pass 2: excitation.  Single 512-thread block (16 waves).
//   GEMM1: h = relu(sq[64,512] @ W1^T + b1)   -> LDS (16 KB)
//   GEMM2: w = sigmoid(h[64,64] @ W2^T + b2)  -> weights[64,512]
// Both via V_WMMA_F32_16X16X4_F32.  Fragment layouts per CDNA5 ISA 7.12.2:
//   A 16x4 f32 : lane(0..15)=M row, VGPR{0,1}=K{0,1}; lanes 16..31 -> K{2,3}
//   B 4x16 f32 : lane(0..15)=N col, VGPR{0,1}=K{0,1}; lanes 16..31 -> K{2,3}
//   C/D 16x16  : VGPR v -> M = v + 8*(lane>=16), N = lane&15
// ---------------------------------------------------------------------------
__global__ __launch_bounds__(512) void se_excite(const float* __restrict__ sq,
                                                 const float* __restrict__ W1,
                                                 const float* __restrict__ b1,
                                                 const float* __restrict__ W2,
                                                 const float* __restrict__ b2,
                                                 float* __restrict__ wt) {
  __shared__ float hbuf[HH * HH];  // 64x64 bottleneck activations

  const int tid   = threadIdx.x;
  const int wave  = tid >> 5;
  const int lane  = tid & 31;
  const int row   = lane & 15;   // M-row (A) / N-col (B,C,D)
  const int khalf = lane >> 4;   // selects K pair {0,1} vs {2,3}

  // ---- GEMM1: 4x4 grid of 16x16 tiles, one tile per wave, K=512 ----------
  {
    const int m0 = (wave >> 2) * 16;
    const int n0 = (wave & 3) * 16;
    v8f acc = {};
    for (int k0 = 0; k0 < CC; k0 += 4) {
      const int ka = k0 + 2 * khalf;
      v2f a, b;
      a.x = sq[(m0 + row) * CC + ka];
      a.y = sq[(m0 + row) * CC + ka + 1];
      // B[k][n] = W1^T[k][n] = W1[n][k], W1 row-major [H,C]
      b.x = W1[(n0 + row) * CC + ka];
      b.y = W1[(n0 + row) * CC + ka + 1];
      acc = __builtin_amdgcn_wmma_f32_16x16x4_f32(false, a, false, b,
                                                  (short)0, acc, false, false);
    }
    const float bias = b1[n0 + row];
#pragma unroll
    for (int v = 0; v < 8; ++v) {
      const int m = m0 + v + 8 * khalf;
      const int n = n0 + row;
      const float r = acc[v] + bias;
      hbuf[m * HH + n] = r > 0.f ? r : 0.f;
    }
  }
  __syncthreads();

  // ---- GEMM2: 4x32 grid of 16x16 tiles, 8 tiles per wave, K=64 -----------
#pragma unroll
  for (int i = 0; i < 8; ++i) {
    const int t  = wave * 8 + i;
    const int m0 = (t >> 5) * 16;   // batch-tile (M=64 -> 4 tiles)
    const int n0 = (t & 31) * 16;   // channel-tile (N=512 -> 32 tiles)
    v8f acc = {};
#pragma unroll
    for (int k0 = 0; k0 < HH; k0 += 4) {
      const int ka = k0 + 2 * khalf;
      v2f a, b;
      a.x = hbuf[(m0 + row) * HH + ka];
      a.y = hbuf[(m0 + row) * HH + ka + 1];
      // B[k][n] = W2^T[k][n] = W2[n][k], W2 row-major [C,H]
      b.x = W2[(n0 + row) * HH + ka];
      b.y = W2[(n0 + row) * HH + ka + 1];
      acc = __builtin_amdgcn_wmma_f32_16x16x4_f32(false, a, false, b,
                                                  (short)0, acc, false, false);
    }
    const float bias = b2[n0 + row];
#pragma unroll
    for (int v = 0; v < 8; ++v) {
      const int m = m0 + v + 8 * khalf;       // batch index
      const int n = n0 + row;                 // channel index
      const float z = acc[v] + bias;
      wt[m * CC + n] = 1.0f / (1.0f + __expf(-z));
    }
  }
}

// ---------------------------------------------------------------------------
// Pass 3: out[b,c,:] = x[b,c,:] * wt[b,c].  One block per row; NT streaming
// 128-bit loads/stores (no reuse; keep the 1 GB stream out of L2).
// ---------------------------------------------------------------------------
__global__ __launch_bounds__(256) void se_scale(const float* __restrict__ x,
                                                const float* __restrict__ wt,
                                                float* __restrict__ out) {
  const int row = blockIdx.x;
  const float w = wt[row];
  const f4* xr = (const f4*)(x + (size_t)row * TT);
  f4* orow     = (f4*)(out + (size_t)row * TT);
  const int tid = threadIdx.x;

#pragma unroll
  for (int i = 0; i < 4; ++i) {
    f4 v = __builtin_nontemporal_load(&xr[tid + i * 256]);
    v *= w;
    __builtin_nontemporal_store(v, &orow[tid + i * 256]);
  }
}

// ---------------------------------------------------------------------------
extern "C" void kernel_launch(void* const* d_in, const int* in_sizes, int n_in,
                              void* d_out, int out_size, void* d_ws,
                              size_t ws_size, hipStream_t stream) {
  const float* x  = (const float*)d_in[0];  // [64,512,4096]
  const float* W1 = (const float*)d_in[1];  // [64,512]
  const float* b1 = (const float*)d_in[2];  // [64]
  const float* W2 = (const float*)d_in[3];  // [512,64]
  const float* b2 = (const float*)d_in[4];  // [512]
  float* out = (float*)d_out;

  float* sq = (float*)d_ws;     // [B*C] squeezed means
  float* wt = sq + BB * CC;     // [B*C] sigmoid channel weights

  se_mean  <<<BB * CC, 256, 0, stream>>>(x, sq);
  se_excite<<<1, 512, 0, stream>>>(sq, W1, b1, W2, b2, wt);
  se_scale <<<BB * CC, 256, 0, stream>>>(x, wt, out);
}